// RelativePositionalEncoding2D_76501957476438
// MI455X (gfx1250) — compile-verified
//
#include <hip/hip_runtime.h>
#include <stdint.h>

// RelativePositionalEncoding2D: out[b,i,j,:] = x[b,i,j,:] + (W[:,bin(i-j)] + bias)
// Pure bandwidth-bound: 256MB in + 256MB out @ 23.3 TB/s ~= 22us roofline.
// Strategy: precompute 63x128 (bin-major) table = W.T + bias into d_ws,
// async-DMA it into LDS per block (CDNA5 global_load_async_to_lds_b128),
// then stream x -> out with non-temporal B128 ops, 8 fully-unrolled
// iterations per thread for deep MLP. One wave == one d-row, so all 32
// lanes share a bin and the LDS row read is conflict-free.

typedef float v4f __attribute__((ext_vector_type(4)));

#define D_MODEL      128
#define NBINS        63                       // 2*32-1
#define LSEQ         512
#define NBATCH       2
#define TABLE_F4     (NBINS * (D_MODEL / 4))  // 2016 float4
#define TABLE_BYTES  (TABLE_F4 * 16)          // 32256 B
#define N4_TOTAL     (NBATCH * LSEQ * LSEQ * (D_MODEL / 4))  // 16,777,216 = 1<<24
#define MAIN_BLOCKS  8192
#define MAIN_THREADS 256
#define GRID_STRIDE  (MAIN_BLOCKS * MAIN_THREADS)            // 1<<21
#define ITERS        (N4_TOTAL / GRID_STRIDE)                // exactly 8

// --- kernel A: table[bin*128 + d] = W[d*63 + bin] + bias[d] (bin-major, bias folded)
__global__ void build_table_kernel(const float* __restrict__ W,
                                   const float* __restrict__ bias,
                                   float* __restrict__ tbl) {
    const int bin = blockIdx.x;    // 0..62
    const int d   = threadIdx.x;   // 0..127
    tbl[bin * D_MODEL + d] = W[d * NBINS + bin] + bias[d];
}

__device__ __forceinline__ unsigned bin_of(unsigned idx4) {
    const unsigned jj = (idx4 >> 5) & (LSEQ - 1);
    const unsigned ii = (idx4 >> 14) & (LSEQ - 1);
    int rel = (int)ii - (int)jj;
    rel = rel < -31 ? -31 : (rel > 31 ? 31 : rel);
    return (unsigned)(rel + 31);
}

__device__ __forceinline__ void stream_body(const v4f* __restrict__ x,
                                            v4f* __restrict__ out,
                                            const v4f* tab) {
    const unsigned idx0 = blockIdx.x * MAIN_THREADS + threadIdx.x;  // < GRID_STRIDE
    const unsigned d4   = idx0 & 31u;

    // Exactly ITERS trips: issue all NT loads / ds loads up-front via full unroll.
    v4f xv[ITERS], tv[ITERS];
#pragma unroll
    for (int it = 0; it < ITERS; ++it) {
        const unsigned idx = idx0 + (unsigned)it * GRID_STRIDE;
        xv[it] = __builtin_nontemporal_load(&x[idx]);
        tv[it] = tab[bin_of(idx) * 32 + d4];   // ds_load_b128, wave-uniform bin
    }
#pragma unroll
    for (int it = 0; it < ITERS; ++it) {
        const unsigned idx = idx0 + (unsigned)it * GRID_STRIDE;
        __builtin_nontemporal_store(xv[it] + tv[it], &out[idx]);
    }
}

// --- kernel B (preferred): async-DMA table into LDS, then NT-stream x -> out
__global__ void __launch_bounds__(MAIN_THREADS)
add_relpos_async(const v4f* __restrict__ x,
                 const v4f* __restrict__ tbl,
                 v4f* __restrict__ out) {
    __shared__ v4f tab[TABLE_F4];   // 32256 B

    // CDNA5 async global->LDS copy (GV mode, SADDR=NULL). Each lane moves 16B.
    // 2016/256 = 7 full passes + one 7-wave partial pass (wave-aligned EXEC).
    for (unsigned e = threadIdx.x; e < TABLE_F4; e += MAIN_THREADS) {
        unsigned lds_addr = (unsigned)(uintptr_t)(&tab[e]);  // flat shared addr[31:0] == LDS offset
        uint64_t gaddr    = (uint64_t)(uintptr_t)(tbl + e);
        asm volatile("global_load_async_to_lds_b128 %0, %1, off"
                     :: "v"(lds_addr), "v"(gaddr) : "memory");
    }
    asm volatile("s_wait_asynccnt 0x0" ::: "memory");
    __syncthreads();

    stream_body(x, out, tab);
}

// --- fallback if d_ws is too small: build table in LDS directly from W/bias
__global__ void __launch_bounds__(MAIN_THREADS)
add_relpos_fallback(const float* __restrict__ W,
                    const float* __restrict__ bias,
                    const v4f* __restrict__ x,
                    v4f* __restrict__ out) {
    __shared__ v4f tab[TABLE_F4];
    float* tabf = (float*)tab;
    for (unsigned e = threadIdx.x; e < NBINS * D_MODEL; e += MAIN_THREADS) {
        const unsigned bin = e >> 7;
        const unsigned d   = e & 127u;
        tabf[e] = W[d * NBINS + bin] + bias[d];
    }
    __syncthreads();

    stream_body(x, out, tab);
}

extern "C" void kernel_launch(void* const* d_in, const int* in_sizes, int n_in,
                              void* d_out, int out_size, void* d_ws, size_t ws_size,
                              hipStream_t stream) {
    const float* x    = (const float*)d_in[0];
    // d_in[1] = idx (int64) is unused by the reference forward
    const float* W    = (const float*)d_in[2];
    const float* bias = (const float*)d_in[3];
    float*       out  = (float*)d_out;

    if (ws_size >= (size_t)TABLE_BYTES) {
        float* tbl = (float*)d_ws;
        build_table_kernel<<<NBINS, D_MODEL, 0, stream>>>(W, bias, tbl);
        add_relpos_async<<<MAIN_BLOCKS, MAIN_THREADS, 0, stream>>>(
            (const v4f*)x, (const v4f*)tbl, (v4f*)out);
    } else {
        add_relpos_fallback<<<MAIN_BLOCKS, MAIN_THREADS, 0, stream>>>(
            W, bias, (const v4f*)x, (v4f*)out);
    }
}